// PSAGE_59657095741761
// MI455X (gfx1250) — compile-verified
//
#include <hip/hip_runtime.h>
#include <math.h>

// ---------------------------------------------------------------------------
// GraphSAGE (3x SAGEConv, mean aggr) for MI455X / gfx1250.
//
// Roofline: per layer ~0.8-1.0 GB edge gather/scatter traffic vs ~3.3 GFLOP
// of GEMM -> memory/atomic bound. Therefore:
//   * scatter uses native no-return global_atomic_add_f32 (inline asm, device
//     scope) into a 25.6MB accumulator that stays resident in the 192MB L2
//   * GEMMs use full-precision V_WMMA_F32_16X16X4_F32 (precision is free,
//     compute is not the bottleneck)
//   * GEMM tiles are staged by the Tensor Data Mover (TENSOR_LOAD_TO_LDS,
//     2D tile descriptor with HW row padding) + s_wait_tensorcnt
// ---------------------------------------------------------------------------

#define DIN 128                      // feature width of x / h1 / h2
typedef __attribute__((ext_vector_type(2))) float        v2f;
typedef __attribute__((ext_vector_type(8))) float        v8f;
typedef __attribute__((ext_vector_type(4))) unsigned int v4u;
typedef __attribute__((ext_vector_type(8))) int          v8i;
typedef __attribute__((ext_vector_type(4))) int          v4i;

// Native f32 global atomic add, no return (STOREcnt-tracked), device scope.
__device__ __forceinline__ void atomic_add_f32(float* p, float v) {
    asm volatile("global_atomic_add_f32 %0, %1, off scope:SCOPE_DEV"
                 :: "v"(p), "v"(v) : "memory");
}

// ---- degree: deg[dst] += 1 ------------------------------------------------
__global__ void sage_deg_kernel(const long long* __restrict__ dst,
                                float* __restrict__ deg, int nE) {
    int e = blockIdx.x * blockDim.x + threadIdx.x;
    if (e < nE) atomic_add_f32(deg + dst[e], 1.0f);
}

// ---- scatter-add: agg[dst] += h[src], one wave per edge, 4 floats/lane ----
__global__ void sage_scatter_kernel(const float* __restrict__ h,
                                    const long long* __restrict__ src,
                                    const long long* __restrict__ dst,
                                    float* __restrict__ agg, int nE) {
    int idx = blockIdx.x * blockDim.x + threadIdx.x;
    int e = idx >> 5;                 // wave handles one edge (32 lanes * 4 = 128)
    int c = (idx & 31) << 2;          // this lane's 4-float column chunk
    if (e >= nE) return;
    long long s = src[e], t = dst[e];
    const float4 v = *reinterpret_cast<const float4*>(h + (size_t)s * DIN + c);
    float* o = agg + (size_t)t * DIN + c;
    atomic_add_f32(o + 0, v.x);
    atomic_add_f32(o + 1, v.y);
    atomic_add_f32(o + 2, v.z);
    atomic_add_f32(o + 3, v.w);
}

// ---- TDM: DMA a 16x128 f32 tile (row stride 128 f32) into LDS -------------
// LDS layout gets +4 dwords of pad per 128-dword row (bank-conflict-free
// b64 A-fragment reads): pad_interval code 6 (=128 dw), pad_amount code 3 (=4 dw).
#if __has_builtin(__builtin_amdgcn_tensor_load_to_lds)
#define HAVE_TDM 1
__device__ __forceinline__ void tdm_load_tile_16x128(const float* gsrc,
                                                     float* lds_dst) {
    unsigned long long ga = (unsigned long long)(uintptr_t)gsrc;
    v4u g0;
    g0[0] = 1u;                                           // count=1, user D#
    g0[1] = (unsigned)(uintptr_t)lds_dst;                 // LDS byte address
    g0[2] = (unsigned)(ga & 0xFFFFFFFFu);                 // global_addr[31:0]
    g0[3] = (unsigned)((ga >> 32) & 0x01FFFFFFu)          // global_addr[56:32]
          | (2u << 30);                                   // type = 2 (image)
    v8i g1;
    g1[0] = (2 << 16)        // data_size = 4 bytes
          | (1 << 20)        // pad_enable
          | (6 << 22)        // pad_interval: 128 dwords
          | (3 << 25);       // pad_amount:   4 dwords
    g1[1] = (128 << 16);     // tensor_dim0[15:0] = 128 (abar_addr = 0)
    g1[2] = (16 << 16);      // tensor_dim0[31:16]=0 | tensor_dim1[15:0] = 16
    g1[3] = (128 << 16);     // tensor_dim1[31:16]=0 | tile_dim0 = 128
    g1[4] = 16;              // tile_dim1 = 16, tile_dim2 = 0
    g1[5] = 128;             // tensor_dim0_stride[31:0] = 128 elements
    g1[6] = 0;               // stride[47:32]=0 | tensor_dim1_stride lo = 0
    g1[7] = 0;
    v4i gz4 = {0, 0, 0, 0};             // groups 2/3 unused (2D tile)
    v8i gz8 = {0, 0, 0, 0, 0, 0, 0, 0}; // trailing group (clang-23 6-arg form)
    __builtin_amdgcn_tensor_load_to_lds(g0, g1, gz4, gz4, gz8, 0);
}
#else
#define HAVE_TDM 0
#endif

// ---- fused normalize + dual GEMM + bias + activation ----------------------
// out[m][n] = act( sum_k (agg[m][k]/max(deg,1)) * Wl[n][k]
//                + sum_k xin[m][k]              * Wr[n][k] + bl[n] )
// Block: 16 rows of nodes; wave w owns output columns [16w, 16w+16).
// A-frag (16x4 f32): lane l -> row l&15, k = 2*(l>>4) + {vgpr0,vgpr1}
// B-frag (4x16):     lane l -> col l&15, k = 2*(l>>4) + {vgpr0,vgpr1}
// C/D (16x16): vgpr r, lane l -> row r + 8*(l>>4), col l&15
template <int D_OUT, int ACT>   // ACT: 0 = ReLU, 1 = tanh
__global__ __launch_bounds__(D_OUT * 2)
void sage_gemm_kernel(const float* __restrict__ agg, const float* __restrict__ deg,
                      const float* __restrict__ xin,
                      const float* __restrict__ Wl, const float* __restrict__ bl,
                      const float* __restrict__ Wr, float* __restrict__ out) {
    constexpr int NT  = D_OUT * 2;    // threads per block (2 per output column)
    constexpr int LDA = DIN + 4;      // padded row stride (matches TDM pad)
    __shared__ float As[16 * LDA];    // raw aggregate tile
    __shared__ float Xs[16 * LDA];    // self-feature tile

    const int tid  = threadIdx.x;
    const int lane = tid & 31;
    const int wave = tid >> 5;
    const int m0   = blockIdx.x * 16;
    const int mloc = lane & 15;
    const int khi  = lane >> 4;       // 0 or 1 -> K sub-offset 2*khi
    const int ncol = wave * 16 + mloc;

#if HAVE_TDM
    if (tid < 32) {                   // one wave issues both DMAs
        tdm_load_tile_16x128(agg + (size_t)m0 * DIN, As);
        tdm_load_tile_16x128(xin + (size_t)m0 * DIN, Xs);
        __builtin_amdgcn_s_wait_tensorcnt(0);
    }
    __syncthreads();
#else
    for (int i = tid; i < 16 * (DIN / 4); i += NT) {
        int row = i / (DIN / 4);
        int col = (i % (DIN / 4)) * 4;
        const float4 va = *reinterpret_cast<const float4*>(
            agg + (size_t)(m0 + row) * DIN + col);
        const float4 vx = *reinterpret_cast<const float4*>(
            xin + (size_t)(m0 + row) * DIN + col);
        float* pa = As + row * LDA + col;
        float* px = Xs + row * LDA + col;
        pa[0] = va.x; pa[1] = va.y; pa[2] = va.z; pa[3] = va.w;
        px[0] = vx.x; px[1] = vx.y; px[2] = vx.z; px[3] = vx.w;
    }
    __syncthreads();
#endif

    // per-lane mean-normalization factor (A row is fixed per lane)
    const float invd = 1.0f / fmaxf(deg[m0 + mloc], 1.0f);

    v8f acc = {};
    const float* WlRow = Wl + (size_t)ncol * DIN;
#pragma unroll 4
    for (int k0 = 0; k0 < DIN; k0 += 4) {
        int ka = k0 + 2 * khi;
        v2f a; a.x = As[mloc * LDA + ka] * invd; a.y = As[mloc * LDA + ka + 1] * invd;
        v2f b; b.x = WlRow[ka];                  b.y = WlRow[ka + 1];
        acc = __builtin_amdgcn_wmma_f32_16x16x4_f32(
            false, a, false, b, (short)0, acc, false, false);
    }

    const float* WrRow = Wr + (size_t)ncol * DIN;
#pragma unroll 4
    for (int k0 = 0; k0 < DIN; k0 += 4) {
        int ka = k0 + 2 * khi;
        v2f a; a.x = Xs[mloc * LDA + ka]; a.y = Xs[mloc * LDA + ka + 1];
        v2f b; b.x = WrRow[ka];           b.y = WrRow[ka + 1];
        acc = __builtin_amdgcn_wmma_f32_16x16x4_f32(
            false, a, false, b, (short)0, acc, false, false);
    }

    const float bias = bl[ncol];
#pragma unroll
    for (int r = 0; r < 8; ++r) {
        int m = m0 + r + 8 * khi;
        float v = acc[r] + bias;
        v = (ACT == 0) ? fmaxf(v, 0.0f) : tanhf(v);
        out[(size_t)m * D_OUT + ncol] = v;
    }
}

// ---------------------------------------------------------------------------
extern "C" void kernel_launch(void* const* d_in, const int* in_sizes, int n_in,
                              void* d_out, int out_size, void* d_ws, size_t ws_size,
                              hipStream_t stream) {
    const int N = in_sizes[0] / DIN;          // 50000
    const int E = in_sizes[1] / 2;            // 800000

    const float*     x   = (const float*)d_in[0];
    const long long* ei  = (const long long*)d_in[1];   // int64 edge_index [2,E]
    const long long* src = ei;
    const long long* dst = ei + E;
    const float* Wl1 = (const float*)d_in[2];
    const float* bl1 = (const float*)d_in[3];
    const float* Wr1 = (const float*)d_in[4];
    const float* Wl2 = (const float*)d_in[5];
    const float* bl2 = (const float*)d_in[6];
    const float* Wr2 = (const float*)d_in[7];
    const float* Wl3 = (const float*)d_in[8];
    const float* bl3 = (const float*)d_in[9];
    const float* Wr3 = (const float*)d_in[10];

    // workspace layout: deg | agg | h1 | h2
    char*  ws  = (char*)d_ws;
    size_t off = 0;
    float* deg = (float*)(ws + off); off += ((size_t)N * sizeof(float) + 255) & ~(size_t)255;
    float* agg = (float*)(ws + off); off += (size_t)N * DIN * sizeof(float);
    float* h1  = (float*)(ws + off); off += (size_t)N * DIN * sizeof(float);
    float* h2  = (float*)(ws + off);

    const size_t aggBytes = (size_t)N * DIN * sizeof(float);
    const int degBlocks  = (E + 255) / 256;
    const int scatBlocks = (int)(((size_t)E * 32 + 255) / 256);
    const int gemmBlocks = N / 16;            // 50000 / 16 = 3125 exact

    // degree (shared by all 3 layers)
    (void)hipMemsetAsync(deg, 0, (size_t)N * sizeof(float), stream);
    sage_deg_kernel<<<degBlocks, 256, 0, stream>>>(dst, deg, E);

    // layer 1: x -> h1 (ReLU)
    (void)hipMemsetAsync(agg, 0, aggBytes, stream);
    sage_scatter_kernel<<<scatBlocks, 256, 0, stream>>>(x, src, dst, agg, E);
    sage_gemm_kernel<128, 0><<<gemmBlocks, 256, 0, stream>>>(agg, deg, x, Wl1, bl1, Wr1, h1);

    // layer 2: h1 -> h2 (ReLU)
    (void)hipMemsetAsync(agg, 0, aggBytes, stream);
    sage_scatter_kernel<<<scatBlocks, 256, 0, stream>>>(h1, src, dst, agg, E);
    sage_gemm_kernel<128, 0><<<gemmBlocks, 256, 0, stream>>>(agg, deg, h1, Wl2, bl2, Wr2, h2);

    // layer 3: h2 -> out (tanh, D_OUT = 32)
    (void)hipMemsetAsync(agg, 0, aggBytes, stream);
    sage_scatter_kernel<<<scatBlocks, 256, 0, stream>>>(h2, src, dst, agg, E);
    sage_gemm_kernel<32, 1><<<gemmBlocks, 64, 0, stream>>>(agg, deg, h2, Wl3, bl3, Wr3,
                                                           (float*)d_out);
}